// GraphSAGE_5171140624748
// MI455X (gfx1250) — compile-verified
//
#include <hip/hip_runtime.h>
#include <hip/hip_bf16.h>

typedef __attribute__((ext_vector_type(2))) float v2f;
typedef __attribute__((ext_vector_type(8))) float v8f;

#define NN 50000
#define NE 800000
#define ROWT (NN / 16)           // 3125 row tiles
#define ROWB ((ROWT + 3) / 4)    // 782 row blocks of 4 tiles
// IN_CH == HID_CH == 128, OUT_CH == 32

// ---------------------------------------------------------------- degree
__global__ void sage_degree(const int* __restrict__ dst, float* __restrict__ cnt) {
    int e = blockIdx.x * blockDim.x + threadIdx.x;
    if (e < NE) unsafeAtomicAdd(&cnt[dst[e]], 1.0f);
}

// cnt[i] -> 1/max(cnt,1)
__global__ void sage_invcnt(float* __restrict__ cnt) {
    int i = blockIdx.x * blockDim.x + threadIdx.x;
    if (i < NN) cnt[i] = 1.0f / fmaxf(cnt[i], 1.0f);
}

// ------------------------------------------------- scatter-add: one wave/edge
// lane l handles channels [4l, 4l+4): float4 gather from feat[src], 4 fp32
// HW atomics into agg[dst]. Feature matrices are L2-resident (25.6MB << 192MB).
__global__ void sage_scatter(const float* __restrict__ feat,
                             const int* __restrict__ src,
                             const int* __restrict__ dst,
                             float* __restrict__ agg) {
    int wave = blockIdx.x * 8 + (threadIdx.x >> 5);
    int lane = threadIdx.x & 31;
    if (wave >= NE) return;
    int s = src[wave];
    int d = dst[wave];
    const float4 v = *(const float4*)(feat + (size_t)s * 128 + lane * 4);
    float* base = agg + (size_t)d * 128 + lane * 4;
    unsafeAtomicAdd(base + 0, v.x);
    unsafeAtomicAdd(base + 1, v.y);
    unsafeAtomicAdd(base + 2, v.z);
    unsafeAtomicAdd(base + 3, v.w);
}

// agg[i][c] *= invcnt[i]  (mean aggregation)
__global__ void sage_scale(float* __restrict__ agg, const float* __restrict__ invc) {
    int i = blockIdx.x * blockDim.x + threadIdx.x;   // NN*128 threads exactly
    agg[i] *= invc[i >> 7];
}

// ------------------------------------------- W pre-pack into K-pair layout
// W [128, ncols] row-major -> Wp[kp][col][2] = { W[2kp][col], W[2kp+1][col] }
// so the WMMA B operand ({row k, row k+1} per lane) is one coalesced b64 load.
__global__ void sage_packW(const float* __restrict__ W, float* __restrict__ Wp,
                           int ncols) {
    int i = blockIdx.x * blockDim.x + threadIdx.x;   // 64*ncols pair slots
    if (i >= 64 * ncols) return;
    int kp  = i / ncols;
    int col = i % ncols;
    Wp[(size_t)i * 2 + 0] = W[(size_t)(2 * kp)     * ncols + col];
    Wp[(size_t)i * 2 + 1] = W[(size_t)(2 * kp + 1) * ncols + col];
}

// ---------------------------------------------------------- fused dual GEMM
// out = A0 @ W0 + A1 @ W1 + bias (+ReLU). One wave: 4 row tiles x 1 col tile.
// V_WMMA_F32_16X16X4_F32 layout (ISA 7.12.2):
//   A 16x4:  lanes 0-15 -> M=lane, VGPR{0,1}=K{0,1}; lanes 16-31 -> K{2,3}
//   B 4x16:  lanes 0-15 -> N=lane, VGPR{0,1}=K{0,1}; lanes 16-31 -> K{2,3}
//   C 16x16: VGPR r -> M=r (lanes 0-15) / M=r+8 (lanes 16-31), N=lane&15
// Ping-pong double-buffered pipeline: two operand sets alternate between
// "being loaded" and "being consumed" -> no rotation movs, WMMA bursts
// overlap the other set's 10 in-flight b64 loads.
__global__ void sage_gemm_wmma(const float* __restrict__ A0,
                               const float* __restrict__ A1,
                               const float* __restrict__ W0p,  // packed
                               const float* __restrict__ W1p,  // packed
                               const float* __restrict__ bias,
                               float* __restrict__ out,
                               int ncols, int do_relu) {
    const int colTiles = ncols >> 4;
    const int totalWaves = ROWB * colTiles;
    int wave = blockIdx.x * 8 + (threadIdx.x >> 5);   // wave-uniform guard:
    if (wave >= totalWaves) return;                   // EXEC all-ones for WMMA
    int rb      = wave / colTiles;
    int colTile = wave % colTiles;
    int lane = threadIdx.x & 31;
    int half = lane >> 4;      // 0: K{0,1}, 1: K{2,3}
    int l16  = lane & 15;
    int col  = colTile * 16 + l16;

    // 4 row tiles, clamped to the last valid tile (dup work, identical stores)
    const float* a0p[4];
    const float* a1p[4];
    int rt[4];
    #pragma unroll
    for (int t = 0; t < 4; ++t) {
        int r = rb * 4 + t;
        rt[t] = (r > ROWT - 1) ? (ROWT - 1) : r;
        int row = rt[t] * 16 + l16;
        a0p[t] = A0 + (size_t)row * 128;
        a1p[t] = A1 + (size_t)row * 128;
    }
    // B base pointers for this lane's (kp, col) walk
    const float* b0p = W0p + ((size_t)half * ncols + col) * 2;
    const float* b1p = W1p + ((size_t)half * ncols + col) * 2;
    const int bstep = 2 * ncols * 2;                 // floats per k-step of 4

    v8f acc[4] = {};
    v2f bb0[2], bb1[2], aa0[2][4], aa1[2][4];

    // load k-step s (k = 4s) into buffer `buf`
    auto loadStage = [&](int buf, int s) {
        int ka = s * 4 + half * 2;
        size_t boff = (size_t)s * bstep;
        bb0[buf] = *(const v2f*)(b0p + boff);
        bb1[buf] = *(const v2f*)(b1p + boff);
        #pragma unroll
        for (int t = 0; t < 4; ++t) {
            aa0[buf][t] = *(const v2f*)(a0p[t] + ka);
            aa1[buf][t] = *(const v2f*)(a1p[t] + ka);
        }
    };
    // 8 WMMAs consuming buffer `buf` (4 independent accumulator chains)
    auto compute = [&](int buf) {
        #pragma unroll
        for (int t = 0; t < 4; ++t) {
            acc[t] = __builtin_amdgcn_wmma_f32_16x16x4_f32(
                false, aa0[buf][t], false, bb0[buf], (short)0, acc[t], false, false);
            acc[t] = __builtin_amdgcn_wmma_f32_16x16x4_f32(
                false, aa1[buf][t], false, bb1[buf], (short)0, acc[t], false, false);
        }
    };

    loadStage(0, 0);                                 // prologue: s = 0
    for (int s = 0; s < 30; s += 2) {                // 15 ping-pong iterations
        loadStage(1, s + 1);
        compute(0);                                  // k-step s
        loadStage(0, s + 2);
        compute(1);                                  // k-step s+1
    }
    loadStage(1, 31);
    compute(0);                                      // k-step 30
    compute(1);                                      // k-step 31

    float bv = bias[col];
    #pragma unroll
    for (int t = 0; t < 4; ++t) {
        int orow = rt[t] * 16 + half * 8;
        #pragma unroll
        for (int r = 0; r < 8; ++r) {
            float v = acc[t][r] + bv;
            if (do_relu) v = fmaxf(v, 0.0f);
            out[(size_t)(orow + r) * ncols + col] = v;
        }
    }
}

// ------------------------------------------- row log_softmax, 32ch == wave32
__global__ void sage_logsoftmax(const float* __restrict__ logits,
                                float* __restrict__ out) {
    int wave = blockIdx.x * 8 + (threadIdx.x >> 5);
    if (wave >= NN) return;
    int lane = threadIdx.x & 31;
    float v = logits[(size_t)wave * 32 + lane];
    float m = v;
    #pragma unroll
    for (int off = 16; off >= 1; off >>= 1) m = fmaxf(m, __shfl_xor(m, off, 32));
    float e = __expf(v - m);
    float s = e;
    #pragma unroll
    for (int off = 16; off >= 1; off >>= 1) s += __shfl_xor(s, off, 32);
    out[(size_t)wave * 32 + lane] = (v - m) - __logf(s);
}

// ---------------------------------------------------------------- launcher
extern "C" void kernel_launch(void* const* d_in, const int* in_sizes, int n_in,
                              void* d_out, int out_size, void* d_ws, size_t ws_size,
                              hipStream_t stream) {
    const float* x   = (const float*)d_in[0];
    const int*  edge = (const int*)d_in[1];     // [2, NE]
    const float* W1l = (const float*)d_in[2];
    const float* W1r = (const float*)d_in[3];
    const float* b1  = (const float*)d_in[4];
    const float* W2l = (const float*)d_in[5];
    const float* W2r = (const float*)d_in[6];
    const float* b2  = (const float*)d_in[7];
    float* out = (float*)d_out;

    const int* src = edge;
    const int* dst = edge + NE;

    // workspace layout (floats):
    // cnt[NN] | agg[NN*128] | h[NN*128] | logits[NN*32] | packed weights
    float* cnt    = (float*)d_ws;
    float* agg    = cnt + NN;
    float* h      = agg + (size_t)NN * 128;
    float* logits = h   + (size_t)NN * 128;
    float* W1lp   = logits + (size_t)NN * 32;
    float* W1rp   = W1lp + 128 * 128;
    float* W2lp   = W1rp + 128 * 128;
    float* W2rp   = W2lp + 128 * 32;

    const size_t aggBytes = (size_t)NN * 128 * sizeof(float);

    // ---- weight pre-pack (K-pair layout for coalesced b64 B loads) ----
    sage_packW<<<(64 * 128 + 255) / 256, 256, 0, stream>>>(W1l, W1lp, 128);
    sage_packW<<<(64 * 128 + 255) / 256, 256, 0, stream>>>(W1r, W1rp, 128);
    sage_packW<<<(64 * 32 + 255) / 256, 256, 0, stream>>>(W2l, W2lp, 32);
    sage_packW<<<(64 * 32 + 255) / 256, 256, 0, stream>>>(W2r, W2rp, 32);

    // ---- layer 1 ----
    hipMemsetAsync(cnt, 0, NN * sizeof(float), stream);
    hipMemsetAsync(agg, 0, aggBytes, stream);
    sage_degree<<<(NE + 255) / 256, 256, 0, stream>>>(dst, cnt);
    sage_invcnt<<<(NN + 255) / 256, 256, 0, stream>>>(cnt);
    sage_scatter<<<NE / 8, 256, 0, stream>>>(x, src, dst, agg);
    sage_scale<<<(NN * 128) / 256, 256, 0, stream>>>(agg, cnt);
    // h = relu(mean @ W1l + x @ W1r + b1): 782 row blocks * 8 col tiles
    sage_gemm_wmma<<<(ROWB * 8 + 7) / 8, 256, 0, stream>>>(agg, x, W1lp, W1rp,
                                                           b1, h, 128, 1);

    // ---- layer 2 ----
    hipMemsetAsync(agg, 0, aggBytes, stream);
    sage_scatter<<<NE / 8, 256, 0, stream>>>(h, src, dst, agg);
    sage_scale<<<(NN * 128) / 256, 256, 0, stream>>>(agg, cnt);
    // logits = mean2 @ W2l + h @ W2r + b2: 782 row blocks * 2 col tiles
    sage_gemm_wmma<<<(ROWB * 2 + 7) / 8, 256, 0, stream>>>(agg, h, W2lp, W2rp,
                                                           b2, logits, 32, 0);

    // ---- log_softmax ----
    sage_logsoftmax<<<(NN + 7) / 8, 256, 0, stream>>>(logits, out);
}